// CVQNNClassifier_74251394613858
// MI455X (gfx1250) — compile-verified
//
#include <hip/hip_runtime.h>
#include <hip/hip_bf16.h>
#include <math.h>

typedef float v2f __attribute__((ext_vector_type(2)));
typedef float v4f __attribute__((ext_vector_type(4)));
typedef float v8f __attribute__((ext_vector_type(8)));

#define NMODE 64
#define XSTRIDE 68   // 64 + 4 pad -> conflict-free ds_load_b64 at 16-lane row stride

// ---------------------------------------------------------------------------
// Setup kernel: evolve each column of the 128x128 symplectic matrix (and the
// displacement vector) independently through the sparse transform chain.
// All transcendentals are precomputed in parallel into LDS tables first, so
// the per-column chain is pure FMAs on broadcast LDS reads.
// ---------------------------------------------------------------------------

template <int START>
__device__ __forceinline__ void bs_pass(float col[128], const float* __restrict__ tc,
                                        const float* __restrict__ ts) {
#pragma unroll
  for (int i = START; i < 63; i += 2) {
    const int j = i + 1;
    const float ct = tc[3 * i], st = ts[3 * i];
    const float cp = tc[3 * i + 1], sp = ts[3 * i + 1];
    const float cpst = cp * st, spst = sp * st;
    const float vi = col[i], vj = col[j];
    const float vni = col[64 + i], vnj = col[64 + j];
    col[i]      = ct * vi - cpst * vj - spst * vnj;
    col[j]      = cpst * vi + ct * vj - spst * vni;
    col[64 + i] = spst * vj + ct * vni - cpst * vnj;
    col[64 + j] = spst * vi + cpst * vni + ct * vnj;
  }
}

__device__ __forceinline__ void rot_squeeze(float col[128], const float* __restrict__ tc,
                                            const float* __restrict__ ts,
                                            const float* __restrict__ em,
                                            const float* __restrict__ ep) {
#pragma unroll
  for (int i = 0; i < 64; ++i) {
    const float ci = (i < 63) ? tc[3 * i + 2] : 1.f;
    const float si = (i < 63) ? ts[3 * i + 2] : 0.f;
    const float vi = col[i], vni = col[64 + i];
    const float r0 = ci * vi - si * vni;
    const float r1 = si * vi + ci * vni;
    col[i]      = em[i] * r0;
    col[64 + i] = ep[i] * r1;
  }
}

__device__ __forceinline__ void apply_layer(float col[128], const float* __restrict__ tc1,
                                            const float* __restrict__ ts1,
                                            const float* __restrict__ tc2,
                                            const float* __restrict__ ts2,
                                            const float* __restrict__ em,
                                            const float* __restrict__ ep) {
  bs_pass<0>(col, tc1, ts1);
  bs_pass<1>(col, tc1, ts1);
  rot_squeeze(col, tc1, ts1, em, ep);
  bs_pass<0>(col, tc2, ts2);
  bs_pass<1>(col, tc2, ts2);
}

// ws layout (floats): [0..2047] W2 (32 x 64, rows >=20 zero)
//                     [2048..2079] bias dsel (rows >=20 zero)
//                     [2080..2095] cov_term (w >=10 zero)
__global__ __launch_bounds__(160) void cvqnn_setup(
    const float* __restrict__ i1a, const float* __restrict__ sqa,
    const float* __restrict__ i2a, const float* __restrict__ da,
    const float* __restrict__ i1b, const float* __restrict__ sqb,
    const float* __restrict__ i2b, const float* __restrict__ db,
    float* __restrict__ ws) {
  // Trig/exp tables: array order 0=i1a 1=i2a 2=i1b 3=i2b (each 189 params).
  __shared__ float tcos[4][192];
  __shared__ float tsin[4][192];
  __shared__ float esq[2][2][64];  // [layer][0=exp(-s),1=exp(+s)][mode]
  __shared__ float rowsS[20][128];
  __shared__ float dsel[20];

  const int tid = threadIdx.x;

  // ---- Phase 0: parallel transcendental evaluation (uniform work split) ----
  for (int idx = tid; idx < 4 * 192; idx += blockDim.x) {
    const int a = idx / 192;
    const int e = idx - a * 192;
    const float* p = (a == 0) ? i1a : (a == 1) ? i2a : (a == 2) ? i1b : i2b;
    const float v = (e < 189) ? p[e] : 0.f;
    tcos[a][e] = cosf(v);
    tsin[a][e] = sinf(v);
  }
  for (int idx = tid; idx < 128; idx += blockDim.x) {
    const int l = idx >> 6;
    const int i = idx & 63;
    const float s = (l ? sqb : sqa)[i];
    esq[l][0][i] = expf(-s);
    esq[l][1][i] = expf(s);
  }
  __syncthreads();

  // ---- Phase 1: per-column symplectic chain (pure FMAs) ----
  const bool active = (tid <= 128);
  const bool isD = (tid == 128);
  const int c = isD ? 0 : tid;

  if (active) {
    float col[128];
#pragma unroll
    for (int r = 0; r < 128; ++r) col[r] = 0.f;
    if (!isD) col[c] = 1.f;

    apply_layer(col, tcos[0], tsin[0], tcos[1], tsin[1], esq[0][0], esq[0][1]);
    if (isD) {
#pragma unroll
      for (int i = 0; i < 64; ++i) col[i] += 2.f * da[i];
    }
    apply_layer(col, tcos[2], tsin[2], tcos[3], tsin[3], esq[1][0], esq[1][1]);
    if (isD) {
#pragma unroll
      for (int i = 0; i < 64; ++i) col[i] += 2.f * db[i];
    }

    if (!isD) {
#pragma unroll
      for (int r = 0; r < 10; ++r) {
        rowsS[r][c] = col[r];
        rowsS[10 + r][c] = col[64 + r];
      }
    } else {
#pragma unroll
      for (int r = 0; r < 10; ++r) {
        dsel[r] = col[r];
        dsel[10 + r] = col[64 + r];
      }
    }
  }
  __syncthreads();

  // ---- Phase 2: emit W2, bias, cov_term to workspace ----
  for (int idx = tid; idx < 2048; idx += blockDim.x) {
    const int r = idx >> 6;
    const int k = idx & 63;
    ws[idx] = (r < 20) ? 2.f * rowsS[r][k] : 0.f;
  }
  if (tid < 32) ws[2048 + tid] = (tid < 20) ? dsel[tid] : 0.f;
  if (tid < 16) {
    float s = 0.f;
    if (tid < 10) {
      for (int cc = 0; cc < 128; ++cc) {
        const float a = rowsS[tid][cc];
        const float b = rowsS[10 + tid][cc];
        s += a * a + b * b;
      }
    }
    ws[2080 + tid] = s;
  }
}

// ---------------------------------------------------------------------------
// Main kernel: (1e6 x 64) @ (64 x 20) + bias via V_WMMA_F32_16X16X4_F32,
// fused epilogue log1p(relu((cov + mx^2 + mp^2)/4 - 0.5)).
// 256 threads = 8 waves; each wave owns one 16-row M tile; 128 rows / block.
// ---------------------------------------------------------------------------
__global__ __launch_bounds__(256) void cvqnn_main(const float* __restrict__ x,
                                                  const float* __restrict__ ws,
                                                  float* __restrict__ out, int B) {
  __shared__ float xs[128 * XSTRIDE];  // x tile, padded
  __shared__ float wl[32 * XSTRIDE];   // weight tile, padded

  const int tid = threadIdx.x;
  const int rowBlock = blockIdx.x * 128;

  // Stage W2 (32x64) into LDS: 512 float4 loads, coalesced (L2-resident).
  for (int q = tid; q < 512; q += 256) {
    const int r = q >> 4;
    const int k4 = (q & 15) << 2;
    v4f v = *(const v4f*)(ws + r * 64 + k4);
    *(v4f*)(wl + r * XSTRIDE + k4) = v;
  }
  // Stage x tile (128x64) into LDS: 2048 float4 loads, coalesced; zero-pad tail.
  for (int q = tid; q < 2048; q += 256) {
    const int r = q >> 4;
    const int k4 = (q & 15) << 2;
    const int row = rowBlock + r;
    v4f v;
    if (row < B) {
      v = *(const v4f*)(x + (size_t)row * 64 + k4);
    } else {
      v = (v4f)(0.f);
    }
    *(v4f*)(xs + r * XSTRIDE + k4) = v;
  }
  __syncthreads();

  const int lane = tid & 31;
  const int wid = tid >> 5;
  const int l15 = lane & 15;
  const int hi = lane >> 4;      // 0: K+0/K+1 half, 1: K+2/K+3 half
  const int hi2 = hi << 1;

  // Bias folded into accumulator init; cov_term per output column.
  const float b0 = ws[2048 + l15];
  const float b1 = ws[2048 + 16 + l15];
  const float ctw = ws[2080 + l15];

  v8f c0, c1;
#pragma unroll
  for (int g = 0; g < 8; ++g) {
    c0[g] = b0;
    c1[g] = b1;
  }

  // A fragment: lane L holds x[row = base + (L&15)][k + 2*(L>=16) .. +1]
  const float* ap = xs + (wid * 16 + l15) * XSTRIDE + hi2;
  // B fragments: lane L holds W2[n = nt*16 + (L&15)][k + 2*(L>=16) .. +1]
  const float* bp0 = wl + l15 * XSTRIDE + hi2;
  const float* bp1 = wl + (16 + l15) * XSTRIDE + hi2;

#pragma unroll
  for (int kb = 0; kb < 16; ++kb) {
    v2f a = *(const v2f*)(ap + 4 * kb);
    v2f w0 = *(const v2f*)(bp0 + 4 * kb);
    v2f w1 = *(const v2f*)(bp1 + 4 * kb);
    c0 = __builtin_amdgcn_wmma_f32_16x16x4_f32(false, a, false, w0, (short)0, c0, false, false);
    c1 = __builtin_amdgcn_wmma_f32_16x16x4_f32(false, a, false, w1, (short)0, c1, false, false);
  }

  // Epilogue. D layout: VGPR g, lane L -> row M = g + 8*(L>=16), col N = L&15.
  // mx[w] = msel[:, w] (w = l15 < 10), mp[w] = msel[:, 10+w]:
  //   w < 6  -> column 10+w of tile0 ; w >= 6 -> column w-6 of tile1.
  const int halfbase = lane & 16;
  const int srcA = halfbase + ((l15 + 10) & 15);
  const int srcB = halfbase + ((l15 >= 6) ? (l15 - 6) : 0);
  const bool wvalid = (l15 < 10);

#pragma unroll
  for (int g = 0; g < 8; ++g) {
    const float mx = c0[g];
    const float mpA = __shfl(c0[g], srcA, 32);
    const float mpB = __shfl(c1[g], srcB, 32);
    const float mp = (l15 < 6) ? mpA : mpB;
    const float nmean = (ctw + mx * mx + mp * mp) * 0.25f - 0.5f;
    const float o = log1pf(fmaxf(nmean, 0.f));
    const int row = rowBlock + wid * 16 + g + (hi ? 8 : 0);
    if (wvalid && row < B) out[(size_t)row * 10 + l15] = o;
  }
}

extern "C" void kernel_launch(void* const* d_in, const int* in_sizes, int n_in,
                              void* d_out, int out_size, void* d_ws, size_t ws_size,
                              hipStream_t stream) {
  const float* x     = (const float*)d_in[0];
  const float* i1a   = (const float*)d_in[1];
  const float* sqa   = (const float*)d_in[2];
  const float* i2a   = (const float*)d_in[3];
  const float* da    = (const float*)d_in[4];
  const float* i1b   = (const float*)d_in[5];
  const float* sqb   = (const float*)d_in[6];
  const float* i2b   = (const float*)d_in[7];
  const float* db    = (const float*)d_in[8];
  float* out = (float*)d_out;
  float* ws  = (float*)d_ws;

  const int B = in_sizes[0] / NMODE;

  cvqnn_setup<<<1, 160, 0, stream>>>(i1a, sqa, i2a, da, i1b, sqb, i2b, db, ws);

  const int blocks = (B + 127) / 128;
  cvqnn_main<<<blocks, 256, 0, stream>>>(x, ws, out, B);
}